// GraphSR_23433341567770
// MI455X (gfx1250) — compile-verified
//
#include <hip/hip_runtime.h>

// ---------------------------------------------------------------------------
// GraphSAGE (2 layers, mean aggr) + linear classifier, fp32, for gfx1250.
// N=20000 nodes, IN=H=256, C=16, E=640000 edges.
// Memory-bound in the edge scatter (~2.6 GB total -> ~110us at 23.3 TB/s);
// GEMMs (10.7 GFLOP) done with native fp32 tensor op V_WMMA_F32_16X16X4_F32.
// ---------------------------------------------------------------------------

typedef __attribute__((ext_vector_type(2))) float v2f;
typedef __attribute__((ext_vector_type(8))) float v8f;

#define NNODES 20000
#define HDIM   256
#define CDIM   16
#define LDSS   260   // padded LDS row stride (dwords): 260%64=4 -> conflict-free frag reads

__device__ __forceinline__ v8f wmma4(v2f a, v2f b, v8f c) {
  // D = A(16x4 f32) x B(4x16 f32) + C(16x16 f32)
  return __builtin_amdgcn_wmma_f32_16x16x4_f32(
      /*neg_a=*/false, a, /*neg_b=*/false, b,
      /*c_mod=*/(short)0, c, /*reuse_a=*/false, /*reuse_b=*/false);
}

// ------------------------------- utility -----------------------------------

__global__ void zero_kernel(float4* __restrict__ p, unsigned n4) {
  unsigned i = blockIdx.x * blockDim.x + threadIdx.x;
  if (i < n4) p[i] = make_float4(0.f, 0.f, 0.f, 0.f);
}

__global__ void deg_kernel(const long long* __restrict__ dst,
                           float* __restrict__ deg, unsigned E) {
  unsigned e = blockIdx.x * blockDim.x + threadIdx.x;
  if (e < E) atomicAdd(&deg[(int)dst[e]], 1.0f);
}

__global__ void invdeg_kernel(float* __restrict__ deg, int n) {
  int i = blockIdx.x * blockDim.x + threadIdx.x;
  if (i < n) deg[i] = 1.0f / fmaxf(deg[i], 1.0f);
}

// One edge handled by 64 threads; each thread moves 4 floats (float4 gather
// from x[src], 4x global_atomic_add_f32 into agg[dst]).
__global__ void scatter_kernel(const long long* __restrict__ src,
                               const long long* __restrict__ dst,
                               const float* __restrict__ x,
                               float* __restrict__ agg, unsigned E) {
  unsigned t = blockIdx.x * blockDim.x + threadIdx.x;
  unsigned e = t >> 6;
  if (e >= E) return;
  unsigned c4 = (t & 63u) << 2;
  const long long s = src[e];
  const long long d = dst[e];
  const float4 v = *(const float4*)(x + (size_t)s * HDIM + c4);
  float* p = agg + (size_t)d * HDIM + c4;
  atomicAdd(p + 0, v.x);
  atomicAdd(p + 1, v.y);
  atomicAdd(p + 2, v.z);
  atomicAdd(p + 3, v.w);
}

// ------------------------- fused SAGE layer GEMM ---------------------------
// out_row_block(16 x 256) = relu?( (agg*invdeg) @ Wl + bias + xin @ Wr )
// FUSE_CLS: additionally z @ Wc + bc -> logits[16 x 16] written to out.
// Block: 128 threads (4 waves). Each wave owns 4 column tiles of 16.
template <bool RELU, bool FUSE_CLS>
__global__ __launch_bounds__(128)
void sage_gemm(const float* __restrict__ agg, const float* __restrict__ invdeg,
               const float* __restrict__ xin,
               const float* __restrict__ Wl, const float* __restrict__ bias,
               const float* __restrict__ Wr,
               const float* __restrict__ Wc, const float* __restrict__ bc,
               float* __restrict__ out) {
  __shared__ float sA[16 * LDSS];                       // scaled-agg rows
  __shared__ float sX[16 * LDSS];                       // self rows
  __shared__ float sZ[FUSE_CLS ? 16 * LDSS : 4];        // z tile (fused path)

  const int row0 = blockIdx.x * 16;
  const int tid  = threadIdx.x;

  // Stage 16x256 A (scaled) and X tiles into LDS: 1024 float4 chunks / 128 thr.
  for (int i = tid; i < 16 * 64; i += 128) {
    const int r  = i >> 6;
    const int c4 = (i & 63) << 2;
    const size_t g = (size_t)(row0 + r) * HDIM + c4;
    const float  s = invdeg[row0 + r];
    const float4 va = *(const float4*)(agg + g);
    const float4 vx = *(const float4*)(xin + g);
    float* pa = &sA[r * LDSS + c4];
    float* px = &sX[r * LDSS + c4];
    pa[0] = va.x * s; pa[1] = va.y * s; pa[2] = va.z * s; pa[3] = va.w * s;
    px[0] = vx.x;     px[1] = vx.y;     px[2] = vx.z;     px[3] = vx.w;
  }
  __syncthreads();

  const int wave = tid >> 5;
  const int lane = tid & 31;
  const int m    = lane & 15;       // row (A) / col (B,C) within tile
  const int kh   = lane >> 4;       // K half-select for 16x16x4 layout

  for (int nt = 0; nt < 4; ++nt) {
    const int ncol = (wave * 4 + nt) * 16;
    const int n    = ncol + m;

    v8f acc;
    const float bv = bias[n];
#pragma unroll
    for (int i = 0; i < 8; ++i) acc[i] = bv;   // bias depends only on N=col

#pragma unroll 4
    for (int k = 0; k < HDIM; k += 4) {
      const int kk = k + 2 * kh;
      const v2f aA = *(const v2f*)(&sA[m * LDSS + kk]);
      const v2f aX = *(const v2f*)(&sX[m * LDSS + kk]);
      v2f bL, bR;
      bL.x = Wl[(size_t)(kk + 0) * HDIM + n];
      bL.y = Wl[(size_t)(kk + 1) * HDIM + n];
      bR.x = Wr[(size_t)(kk + 0) * HDIM + n];
      bR.y = Wr[(size_t)(kk + 1) * HDIM + n];
      acc = wmma4(aA, bL, acc);
      acc = wmma4(aX, bR, acc);
    }

    if (!FUSE_CLS) {
#pragma unroll
      for (int r = 0; r < 8; ++r) {
        const int M = r + 8 * kh;
        float v = acc[r];
        if (RELU) v = fmaxf(v, 0.f);
        out[(size_t)(row0 + M) * HDIM + n] = v;
      }
    } else {
#pragma unroll
      for (int r = 0; r < 8; ++r) {
        const int M = r + 8 * kh;
        sZ[M * LDSS + n] = acc[r];
      }
    }
  }

  if (FUSE_CLS) {
    __syncthreads();
    if (wave == 0) {   // wave-uniform branch: EXEC all-ones inside
      v8f acc;
      const float bv = bc[m];
#pragma unroll
      for (int i = 0; i < 8; ++i) acc[i] = bv;
#pragma unroll 4
      for (int k = 0; k < HDIM; k += 4) {
        const int kk = k + 2 * kh;
        const v2f aZ = *(const v2f*)(&sZ[m * LDSS + kk]);
        v2f bC;
        bC.x = Wc[(size_t)(kk + 0) * CDIM + m];
        bC.y = Wc[(size_t)(kk + 1) * CDIM + m];
        acc = wmma4(aZ, bC, acc);
      }
#pragma unroll
      for (int r = 0; r < 8; ++r) {
        const int M = r + 8 * kh;
        out[(size_t)(row0 + M) * CDIM + m] = acc[r];
      }
    }
  }
}

// ------------------------------- launcher ----------------------------------

extern "C" void kernel_launch(void* const* d_in, const int* in_sizes, int n_in,
                              void* d_out, int out_size, void* d_ws, size_t ws_size,
                              hipStream_t stream) {
  const float*     x   = (const float*)d_in[0];
  const long long* ei  = (const long long*)d_in[1];   // int64 [2, E]
  const float*     W1l = (const float*)d_in[2];
  const float*     b1  = (const float*)d_in[3];
  const float*     W1r = (const float*)d_in[4];
  const float*     W2l = (const float*)d_in[5];
  const float*     b2  = (const float*)d_in[6];
  const float*     W2r = (const float*)d_in[7];
  const float*     Wc  = (const float*)d_in[8];
  const float*     bc  = (const float*)d_in[9];
  float*           out = (float*)d_out;

  const unsigned E = (unsigned)(in_sizes[1] / 2);
  const long long* src = ei;
  const long long* dst = ei + E;

  // Workspace layout (floats): deg[32768 pad] | agg1[N*H] | agg2[N*H] | h[N*H]
  float* ws = (float*)d_ws;
  const size_t NH = (size_t)NNODES * HDIM;
  float* deg  = ws;
  float* agg1 = ws + 32768;
  float* agg2 = agg1 + NH;
  float* h    = agg2 + NH;

  // Zero deg + agg1 + agg2 (h is fully overwritten).
  const unsigned zn4 = (unsigned)((32768 + 2 * NH) / 4);
  zero_kernel<<<(zn4 + 255) / 256, 256, 0, stream>>>((float4*)ws, zn4);

  // Degrees -> inverse degrees (shared by both layers).
  deg_kernel<<<(E + 255) / 256, 256, 0, stream>>>(dst, deg, E);
  invdeg_kernel<<<(NNODES + 255) / 256, 256, 0, stream>>>(deg, NNODES);

  const unsigned sthreads = E * 64u;                 // 64 threads per edge
  const unsigned sblocks  = (sthreads + 255) / 256;

  // Layer 1: aggregate x, then h = relu(agg1*inv @ W1l + b1 + x @ W1r)
  scatter_kernel<<<sblocks, 256, 0, stream>>>(src, dst, x, agg1, E);
  sage_gemm<true, false><<<NNODES / 16, 128, 0, stream>>>(
      agg1, deg, x, W1l, b1, W1r, nullptr, nullptr, h);

  // Layer 2 + classifier fused: logits = (agg2*inv @ W2l + b2 + h @ W2r) @ Wc + bc
  scatter_kernel<<<sblocks, 256, 0, stream>>>(src, dst, h, agg2, E);
  sage_gemm<false, true><<<NNODES / 16, 128, 0, stream>>>(
      agg2, deg, h, W2l, b2, W2r, Wc, bc, out);
}